// VisionMambaCPUFull_53309134078224
// MI455X (gfx1250) — compile-verified
//
#include <hip/hip_runtime.h>
#include <hip/hip_bf16.h>
#include <math.h>

typedef __attribute__((ext_vector_type(2))) float v2f;
typedef __attribute__((ext_vector_type(8))) float v8f;

#define BATCH 32
#define SEQL  577
#define DMODEL 192
#define DINNER 384
#define DSTATE 8
#define DCONV 4
#define DTRANK 12
#define NTOK (BATCH * SEQL)          // 18464, divisible by 16
#define NXZ  (2 * DINNER)            // 768
#define NCHUNK 8
#define LCHUNK 73                    // ceil(577/8); last chunk = 66 steps

// ---------------------------------------------------------------------------
// In-projection GEMM: xz[r, n] = sum_k X[srcRow(r), k] * in_w[n, k]
// One wave computes a 16x64 strip (4 accumulators) of V_WMMA_F32_16X16X4_F32,
// reusing each A fetch across 4 N-tiles. rev=1 flips l within each batch.
// ---------------------------------------------------------------------------
__global__ void gemm_in_kernel(const float* __restrict__ x,
                               const float* __restrict__ in_w,  // (768, 192)
                               float* __restrict__ xz,          // (NTOK, 768)
                               int rev)
{
    const int K = DMODEL;                     // 192
    const int N = NXZ;                        // 768
    const int NT = 4;
    const int grpN = N / (16 * NT);           // 12
    int wave = blockIdx.x * (blockDim.x >> 5) + (threadIdx.x >> 5);
    int lane = threadIdx.x & 31;
    int g = wave % grpN;
    int tileM = wave / grpN;
    if (tileM >= NTOK / 16) return;           // wave-uniform

    int half = lane >> 4;                     // 0: K pair {0,1}; 1: {2,3}
    int l16  = lane & 15;

    int m = tileM * 16 + l16;
    int b = m / SEQL, l = m % SEQL;
    int srcRow = rev ? (b * SEQL + (SEQL - 1 - l)) : m;
    const float* arow = x + (size_t)srcRow * K;

    const float* brow[NT];
    #pragma unroll
    for (int j = 0; j < NT; ++j)
        brow[j] = in_w + (size_t)((g * NT + j) * 16 + l16) * K;

    v8f acc[NT];
    #pragma unroll
    for (int j = 0; j < NT; ++j) acc[j] = {};

    for (int k = 0; k < K; k += 4) {
        v2f av;
        av[0] = arow[k + half * 2 + 0];
        av[1] = arow[k + half * 2 + 1];
        #pragma unroll
        for (int j = 0; j < NT; ++j) {
            v2f bv;
            bv[0] = brow[j][k + half * 2 + 0];
            bv[1] = brow[j][k + half * 2 + 1];
            acc[j] = __builtin_amdgcn_wmma_f32_16x16x4_f32(
                false, av, false, bv, (short)0, acc[j], false, false);
        }
    }

    #pragma unroll
    for (int j = 0; j < NT; ++j) {
        int col = (g * NT + j) * 16 + l16;
        #pragma unroll
        for (int i = 0; i < 8; ++i) {
            int row = tileM * 16 + i + half * 8;
            xz[(size_t)row * N + col] = acc[j][i];
        }
    }
}

// ---------------------------------------------------------------------------
// Depthwise causal conv (k=4) + bias + SiLU on the x_part half of xz.
// ---------------------------------------------------------------------------
__global__ void conv_silu_kernel(const float* __restrict__ xz,
                                 const float* __restrict__ conv_w,  // (384,1,4)
                                 const float* __restrict__ conv_b,  // (384)
                                 float* __restrict__ xc)            // (NTOK,384)
{
    int idx = blockIdx.x * blockDim.x + threadIdx.x;   // exact grid
    int c = idx % DINNER;
    int r = idx / DINNER;
    int b = r / SEQL, l = r % SEQL;
    float s = conv_b[c];
    #pragma unroll
    for (int j = 0; j < DCONV; ++j) {
        int ll = l - (DCONV - 1) + j;
        if (ll >= 0)
            s += xz[((size_t)(b * SEQL + ll)) * NXZ + c] * conv_w[c * DCONV + j];
    }
    float sig = 1.0f / (1.0f + __expf(-s));
    xc[(size_t)r * DINNER + c] = s * sig;
}

// ---------------------------------------------------------------------------
// Per-token: dt_bc = xc @ xproj_w.T (28), tanh on B/C terms,
// dt = softplus(dt_part @ dt_w.T + dt_b) + 1e-4.  Block = 1 token, 384 thr.
// ---------------------------------------------------------------------------
__global__ void dtproj_kernel(const float* __restrict__ xc,
                              const float* __restrict__ xproj_w, // (28, 384)
                              const float* __restrict__ dt_w,    // (384, 12)
                              const float* __restrict__ dt_b,    // (384)
                              float* __restrict__ dt_out,        // (NTOK, 384)
                              float* __restrict__ bc_out)        // (NTOK, 16)
{
    __shared__ float sx[DINNER];
    __shared__ float sbc[DTRANK + 2 * DSTATE];
    int tok = blockIdx.x;
    int t = threadIdx.x;
    sx[t] = xc[(size_t)tok * DINNER + t];
    __syncthreads();
    if (t < DTRANK + 2 * DSTATE) {
        const float* w = xproj_w + t * DINNER;
        float s = 0.0f;
        for (int k = 0; k < DINNER; ++k) s += sx[k] * w[k];
        if (t >= DTRANK) {
            s = tanhf(s);
            bc_out[(size_t)tok * 16 + (t - DTRANK)] = s;
        }
        sbc[t] = s;
    }
    __syncthreads();
    const float* w = dt_w + t * DTRANK;
    float s = dt_b[t];
    #pragma unroll
    for (int k = 0; k < DTRANK; ++k) s += sbc[k] * w[k];
    float sp = (s > 30.0f) ? s : log1pf(__expf(s));
    dt_out[(size_t)tok * DINNER + t] = sp + 1e-4f;
}

// ---------------------------------------------------------------------------
// Chunked selective scan, 3 phases. state update: s = da*s + (1-da)*b*x.
// Phase 1: per (b, chunk, c) compute chunk decay product P and zero-init
//          partial state S0 for each of the 8 states.
// ---------------------------------------------------------------------------
__global__ void scan_phase1(const float* __restrict__ xc,
                            const float* __restrict__ dt,
                            const float* __restrict__ bc,     // (NTOK, 16)
                            const float* __restrict__ A_log,  // (384, 8)
                            float* __restrict__ Pbuf,         // (B,CH,384,8)
                            float* __restrict__ Sbuf)
{
    int c  = threadIdx.x;                 // 0..383
    int b  = blockIdx.x / NCHUNK;
    int ch = blockIdx.x % NCHUNK;
    int l0 = ch * LCHUNK;
    int l1 = l0 + LCHUNK; if (l1 > SEQL) l1 = SEQL;

    float a[DSTATE], P[DSTATE], S[DSTATE];
    #pragma unroll
    for (int s = 0; s < DSTATE; ++s) {
        a[s] = -__expf(A_log[c * DSTATE + s]);
        P[s] = 1.0f; S[s] = 0.0f;
    }
    for (int l = l0; l < l1; ++l) {
        size_t r = (size_t)b * SEQL + l;
        float xt  = xc[r * DINNER + c];
        float dtt = dt[r * DINNER + c];
        #pragma unroll
        for (int s = 0; s < DSTATE; ++s) {
            float da = __expf(dtt * a[s]);
            S[s] = da * S[s] + (1.0f - da) * bc[r * 16 + s] * xt;
            P[s] *= da;
        }
    }
    size_t idx = (((size_t)b * NCHUNK + ch) * DINNER + c) * DSTATE;
    #pragma unroll
    for (int s = 0; s < DSTATE; ++s) { Pbuf[idx + s] = P[s]; Sbuf[idx + s] = S[s]; }
}

// Phase 2: propagate chunk-boundary states serially over 8 chunks.
// One thread per (b, c, s): 32*384*8 = 98304 = 384 blocks * 256.
__global__ void scan_phase2(const float* __restrict__ Pbuf,
                            const float* __restrict__ Sbuf,
                            float* __restrict__ Ibuf)
{
    int t = blockIdx.x * blockDim.x + threadIdx.x;
    int s = t % DSTATE;
    int c = (t / DSTATE) % DINNER;
    int b = t / (DSTATE * DINNER);
    float st = 0.0f;
    for (int ch = 0; ch < NCHUNK; ++ch) {
        size_t idx = (((size_t)b * NCHUNK + ch) * DINNER + c) * DSTATE + s;
        Ibuf[idx] = st;
        st = Pbuf[idx] * st + Sbuf[idx];
    }
}

// Phase 3: re-run each chunk from its true initial state, produce gated
// output. bw direction (rev=1) writes output un-flipped so the out-proj
// GEMM is uniform.
__global__ void scan_phase3(const float* __restrict__ xc,
                            const float* __restrict__ dt,
                            const float* __restrict__ bc,
                            const float* __restrict__ A_log,
                            const float* __restrict__ Dskip,  // (384)
                            const float* __restrict__ xz,     // z half for gate
                            const float* __restrict__ Ibuf,
                            float* __restrict__ yg,           // (NTOK, 384)
                            int rev)
{
    int c  = threadIdx.x;
    int b  = blockIdx.x / NCHUNK;
    int ch = blockIdx.x % NCHUNK;
    int l0 = ch * LCHUNK;
    int l1 = l0 + LCHUNK; if (l1 > SEQL) l1 = SEQL;

    float a[DSTATE], st[DSTATE];
    size_t idx = (((size_t)b * NCHUNK + ch) * DINNER + c) * DSTATE;
    #pragma unroll
    for (int s = 0; s < DSTATE; ++s) {
        a[s]  = -__expf(A_log[c * DSTATE + s]);
        st[s] = Ibuf[idx + s];
    }
    float Dc = Dskip[c];

    for (int l = l0; l < l1; ++l) {
        size_t r = (size_t)b * SEQL + l;
        float xt  = xc[r * DINNER + c];
        float dtt = dt[r * DINNER + c];
        float y = Dc * xt;
        #pragma unroll
        for (int s = 0; s < DSTATE; ++s) {
            float da = __expf(dtt * a[s]);
            st[s] = da * st[s] + (1.0f - da) * bc[r * 16 + s] * xt;
            y += st[s] * bc[r * 16 + DSTATE + s];
        }
        float z = xz[r * NXZ + DINNER + c];
        float g = z / (1.0f + __expf(-z));
        size_t ro = rev ? ((size_t)b * SEQL + (SEQL - 1 - l)) : r;
        yg[ro * DINNER + c] = y * g;
    }
}

// ---------------------------------------------------------------------------
// Out-projection: out = 0.5 * (Yfw @ Wfw^T + Ybw @ Wbw^T).
// One wave computes a 16x64 strip (4 accumulators), both directions fused.
// ---------------------------------------------------------------------------
__global__ void gemm_out_kernel(const float* __restrict__ y0,
                                const float* __restrict__ y1,
                                const float* __restrict__ w0,   // (192, 384)
                                const float* __restrict__ w1,   // (192, 384)
                                float* __restrict__ out)        // (NTOK, 192)
{
    const int K = DINNER;                     // 384
    const int N = DMODEL;                     // 192
    const int NT = 4;
    const int grpN = N / (16 * NT);           // 3
    int wave = blockIdx.x * (blockDim.x >> 5) + (threadIdx.x >> 5);
    int lane = threadIdx.x & 31;
    int g = wave % grpN;
    int tileM = wave / grpN;
    if (tileM >= NTOK / 16) return;           // wave-uniform (covers tail waves)

    int half = lane >> 4;
    int l16  = lane & 15;
    int m = tileM * 16 + l16;

    const float* a0 = y0 + (size_t)m * K;
    const float* a1 = y1 + (size_t)m * K;
    const float* b0[NT];
    const float* b1[NT];
    #pragma unroll
    for (int j = 0; j < NT; ++j) {
        b0[j] = w0 + (size_t)((g * NT + j) * 16 + l16) * K;
        b1[j] = w1 + (size_t)((g * NT + j) * 16 + l16) * K;
    }

    v8f acc[NT];
    #pragma unroll
    for (int j = 0; j < NT; ++j) acc[j] = {};

    for (int k = 0; k < K; k += 4) {
        v2f av;
        av[0] = a0[k + half * 2 + 0];
        av[1] = a0[k + half * 2 + 1];
        #pragma unroll
        for (int j = 0; j < NT; ++j) {
            v2f bv;
            bv[0] = b0[j][k + half * 2 + 0];
            bv[1] = b0[j][k + half * 2 + 1];
            acc[j] = __builtin_amdgcn_wmma_f32_16x16x4_f32(
                false, av, false, bv, (short)0, acc[j], false, false);
        }
    }
    for (int k = 0; k < K; k += 4) {
        v2f av;
        av[0] = a1[k + half * 2 + 0];
        av[1] = a1[k + half * 2 + 1];
        #pragma unroll
        for (int j = 0; j < NT; ++j) {
            v2f bv;
            bv[0] = b1[j][k + half * 2 + 0];
            bv[1] = b1[j][k + half * 2 + 1];
            acc[j] = __builtin_amdgcn_wmma_f32_16x16x4_f32(
                false, av, false, bv, (short)0, acc[j], false, false);
        }
    }

    #pragma unroll
    for (int j = 0; j < NT; ++j) {
        int col = (g * NT + j) * 16 + l16;
        #pragma unroll
        for (int i = 0; i < 8; ++i) {
            int row = tileM * 16 + i + half * 8;
            out[(size_t)row * N + col] = 0.5f * acc[j][i];
        }
    }
}

// ---------------------------------------------------------------------------
extern "C" void kernel_launch(void* const* d_in, const int* in_sizes, int n_in,
                              void* d_out, int out_size, void* d_ws, size_t ws_size,
                              hipStream_t stream) {
    const float* x = (const float*)d_in[0];
    const float* in_w[2]    = { (const float*)d_in[1],  (const float*)d_in[10] };
    const float* conv_w[2]  = { (const float*)d_in[2],  (const float*)d_in[11] };
    const float* conv_b[2]  = { (const float*)d_in[3],  (const float*)d_in[12] };
    const float* xproj_w[2] = { (const float*)d_in[4],  (const float*)d_in[13] };
    const float* dt_w[2]    = { (const float*)d_in[5],  (const float*)d_in[14] };
    const float* dt_b[2]    = { (const float*)d_in[6],  (const float*)d_in[15] };
    const float* A_log[2]   = { (const float*)d_in[7],  (const float*)d_in[16] };
    const float* Dskip[2]   = { (const float*)d_in[8],  (const float*)d_in[17] };
    const float* out_w[2]   = { (const float*)d_in[9],  (const float*)d_in[18] };
    float* out = (float*)d_out;

    // workspace layout (floats)
    float* ws = (float*)d_ws;
    size_t off = 0;
    float *xz[2], *xc[2], *dt[2], *bc[2], *yg[2], *P[2], *S[2], *I[2];
    const size_t CHSZ = (size_t)BATCH * NCHUNK * DINNER * DSTATE;   // 786432
    for (int d = 0; d < 2; ++d) { xz[d] = ws + off; off += (size_t)NTOK * NXZ; }
    for (int d = 0; d < 2; ++d) { xc[d] = ws + off; off += (size_t)NTOK * DINNER; }
    for (int d = 0; d < 2; ++d) { dt[d] = ws + off; off += (size_t)NTOK * DINNER; }
    for (int d = 0; d < 2; ++d) { bc[d] = ws + off; off += (size_t)NTOK * 16; }
    for (int d = 0; d < 2; ++d) { yg[d] = ws + off; off += (size_t)NTOK * DINNER; }
    for (int d = 0; d < 2; ++d) { P[d]  = ws + off; off += CHSZ; }
    for (int d = 0; d < 2; ++d) { S[d]  = ws + off; off += CHSZ; }
    for (int d = 0; d < 2; ++d) { I[d]  = ws + off; off += CHSZ; }

    for (int d = 0; d < 2; ++d) {
        // in-projection: 1154 M-tiles * 12 N-groups = 13848 waves = 1731 blocks
        gemm_in_kernel<<<dim3(1731), 256, 0, stream>>>(x, in_w[d], xz[d], d);

        // conv + silu: NTOK*384 = 7,090,176 = 27696 * 256 (exact)
        conv_silu_kernel<<<dim3(27696), 256, 0, stream>>>(xz[d], conv_w[d], conv_b[d], xc[d]);

        // dt projection: one block per token
        dtproj_kernel<<<dim3(NTOK), DINNER, 0, stream>>>(xc[d], xproj_w[d], dt_w[d],
                                                         dt_b[d], dt[d], bc[d]);

        // chunked scan
        scan_phase1<<<dim3(BATCH * NCHUNK), DINNER, 0, stream>>>(xc[d], dt[d], bc[d],
                                                                 A_log[d], P[d], S[d]);
        scan_phase2<<<dim3(384), 256, 0, stream>>>(P[d], S[d], I[d]);
        scan_phase3<<<dim3(BATCH * NCHUNK), DINNER, 0, stream>>>(xc[d], dt[d], bc[d],
                                                                 A_log[d], Dskip[d], xz[d],
                                                                 I[d], yg[d], d);
    }

    // out-projection: 1154 M-tiles * 3 N-groups = 3462 waves -> 433 blocks (guarded)
    gemm_out_kernel<<<dim3(433), 256, 0, stream>>>(yg[0], yg[1], out_w[0], out_w[1], out);
}